// GatedCommunicationModule_80659485819618
// MI455X (gfx1250) — compile-verified
//
#include <hip/hip_runtime.h>
#include <hip/hip_bf16.h>
#include <math.h>

// GatedCommunicationModule on MI455X (gfx1250, wave32, WMMA).
// B=256, N=64, D=128, H=64. One workgroup (4 waves / 128 thr) per batch elem.
// All GEMMs use V_WMMA_F32_16X16X4_F32 (exact f32). B matrices are staged
// TRANSPOSED in LDS so both A and B operands are single 8B-aligned b64 loads
// into even VGPR pairs (no marshaling movs). All WMMA-operand matrices use
// ld = 66 floats: bank = (2*lane16 + k) -> conflict-free, offsets stay 8B
// aligned. Non-factorable gate reduction sigmoid(sum_h relu(gi+gj)*w2) is
// VALU with broadcast gi reads and conflict-free gj reads.

typedef __attribute__((ext_vector_type(2))) float v2f;
typedef __attribute__((ext_vector_type(8))) float v8f;

#define BATCH 256
#define N_DIM 64
#define D_DIM 128
#define H_DIM 64
#define LDP   66            // padded leading dim for 64-wide matrices

// ---- LDS: three 64x66 regions + w2 ----
// R0 [    0, 4224)  obs col-half -> enc -> tmp
// R1 [ 4224, 8448)  gi -> gates
// R2 [ 8448,12672)  encWT halves -> w_iT -> w_jT -> gj -> mw1T -> mw2T -> msgT
// W2 [12672,12736)  gate_w2

__device__ inline v8f wmma_f32x4(v2f a, v2f b, v8f c) {
  return __builtin_amdgcn_wmma_f32_16x16x4_f32(
      /*neg_a=*/false, a, /*neg_b=*/false, b,
      /*c_mod=*/(short)0, c, /*reuse_a=*/false, /*reuse_b=*/false);
}

__device__ inline void init_acc(v8f acc[4], const float* __restrict__ bias, int l16) {
#pragma unroll
  for (int t = 0; t < 4; ++t) {
    float bv = bias ? bias[t * 16 + l16] : 0.0f;
#pragma unroll
    for (int r = 0; r < 8; ++r) acc[t][r] = bv;
  }
}

// acc += A[rowBase..rowBase+16, 0..K) * B(K x 64), with B stored TRANSPOSED:
// BT[n*ldbt + k]. Both operands load as one 8B-aligned v2f (ds_load_b64).
__device__ inline void gemm_accT(const float* __restrict__ A, int lda,
                                 const float* __restrict__ BT, int ldbt,
                                 int K, v8f acc[4], int lane, int rowBase) {
  const int half = lane >> 4;
  const int l16  = lane & 15;
  for (int k0 = 0; k0 < K; k0 += 4) {
    const int ka = k0 + 2 * half;                       // even -> 8B aligned
    v2f a = *(const v2f*)(A + (rowBase + l16) * lda + ka);
#pragma unroll
    for (int t = 0; t < 4; ++t) {
      v2f b = *(const v2f*)(BT + (t * 16 + l16) * ldbt + ka);
      acc[t] = wmma_f32x4(a, b, acc[t]);
    }
  }
}

// Write 16x64 stripe of C/D accumulators row-major (ld = ldd).
__device__ inline void store_stripe(float* __restrict__ D, int ldd,
                                    const v8f acc[4], int lane, int rowBase,
                                    bool do_relu) {
  const int half = lane >> 4;
  const int l16  = lane & 15;
#pragma unroll
  for (int t = 0; t < 4; ++t)
#pragma unroll
    for (int r = 0; r < 8; ++r) {
      float v = acc[t][r];
      if (do_relu) v = fmaxf(v, 0.0f);
      D[(rowBase + r + 8 * half) * ldd + t * 16 + l16] = v;
    }
}

// Write 16x64 stripe TRANSPOSED: DT[col*ldt + row] (for use as next B operand).
__device__ inline void store_stripeT(float* __restrict__ DT, int ldt,
                                     const v8f acc[4], int lane, int rowBase) {
  const int half = lane >> 4;
  const int l16  = lane & 15;
#pragma unroll
  for (int t = 0; t < 4; ++t)
#pragma unroll
    for (int r = 0; r < 8; ++r)
      DT[(t * 16 + l16) * ldt + rowBase + r + 8 * half] = acc[t][r];
}

// Transposed 64x64 weight stage: dst[n*LDP + k] = src[(k+koff)*64 + n].
// Global reads coalesced over n; LDS writes stride LDP -> banks 2n+k (clean).
__device__ inline void copy_w_T(float* __restrict__ dst,
                                const float* __restrict__ src,
                                int koff, int tid) {
  for (int s = tid; s < 64 * 64; s += 128) {
    const int k = s >> 6;
    const int n = s & 63;
    dst[n * LDP + k] = src[(k + koff) * H_DIM + n];
  }
}

// Stage a 64-column half of obs[b] row-major with ld = LDP (float2 granules).
__device__ inline void copy_obs_half(float* __restrict__ dst,
                                     const float* __restrict__ src,
                                     int coff, int tid) {
  for (int s = tid; s < 64 * 32; s += 128) {      // 64 rows x 32 float2
    const int r  = s >> 5;
    const int c2 = (s & 31) * 2;
    *(float2*)(dst + r * LDP + c2) =
        *(const float2*)(src + r * D_DIM + coff + c2);
  }
}

__global__ __launch_bounds__(128) void gated_comm_kernel(
    const float* __restrict__ obs, const float* __restrict__ enc_w,
    const float* __restrict__ enc_b, const float* __restrict__ gate_w1,
    const float* __restrict__ gate_b1, const float* __restrict__ gate_w2,
    const float* __restrict__ gate_b2, const float* __restrict__ msg_w1,
    const float* __restrict__ msg_b1, const float* __restrict__ msg_w2,
    const float* __restrict__ msg_b2, float* __restrict__ out) {
  __shared__ float smem[3 * 64 * LDP + 64];

  const int b       = blockIdx.x;
  const int tid     = threadIdx.x;
  const int lane    = tid & 31;
  const int wave    = tid >> 5;     // 0..3
  const int l16     = lane & 15;
  const int rowBase = wave * 16;

  float* R0  = smem;                  // obs half -> enc -> tmp
  float* R1  = smem + 64 * LDP;       // gi -> gates
  float* R2  = smem + 2 * 64 * LDP;   // rotating B operand (transposed)
  float* sW2 = smem + 3 * 64 * LDP;   // gate_w2

  const float* obsb = obs + (size_t)b * (N_DIM * D_DIM);
  v8f acc[4];

  // ---- enc = relu(obs @ enc_w + enc_b), K=128 split into two K=64 passes ----
  copy_obs_half(R0, obsb, 0, tid);
  copy_w_T(R2, enc_w, 0, tid);
  __syncthreads();
  init_acc(acc, enc_b, l16);
  gemm_accT(R0, LDP, R2, LDP, 64, acc, lane, rowBase);
  __syncthreads();
  copy_obs_half(R0, obsb, 64, tid);
  copy_w_T(R2, enc_w, 64, tid);
  __syncthreads();
  gemm_accT(R0, LDP, R2, LDP, 64, acc, lane, rowBase);
  __syncthreads();
  store_stripe(R0, LDP, acc, lane, rowBase, /*relu=*/true);   // enc over obs
  copy_w_T(R2, gate_w1, 0, tid);                              // w_i (rows 0..63)
  __syncthreads();

  // ---- gi = enc @ w_i ----
  init_acc(acc, nullptr, l16);
  gemm_accT(R0, LDP, R2, LDP, H_DIM, acc, lane, rowBase);
  store_stripe(R1, LDP, acc, lane, rowBase, false);           // R1 was free
  __syncthreads();
  copy_w_T(R2, gate_w1, 64, tid);                             // w_j (rows 64..127)
  __syncthreads();

  // ---- gj = enc @ w_j + gate_b1 ----
  init_acc(acc, gate_b1, l16);
  gemm_accT(R0, LDP, R2, LDP, H_DIM, acc, lane, rowBase);
  __syncthreads();                                  // all waves done with w_jT
  store_stripe(R2, LDP, acc, lane, rowBase, false); // gj over w_jT
  if (tid < H_DIM) sW2[tid] = gate_w2[tid];
  __syncthreads();

  // ---- gates[i,j] = sigmoid(sum_h relu(gi[i,h]+gj[j,h])*w2[h] + b2) ----
  // Wave owns 16 rows i; lanes sweep j = lane, lane+32. gi reads broadcast
  // (same address); gj reads hit banks (2j+h) -> conflict-free.
  const float gb2 = gate_b2[0];
  float gv[32];
#pragma unroll 1
  for (int ii = 0; ii < 16; ++ii) {
    const int i = rowBase + ii;
    float s0 = 0.0f, s1 = 0.0f;
    for (int h = 0; h < H_DIM; ++h) {
      const float gih = R1[i * LDP + h];
      const float w2h = sW2[h];
      s0 += fmaxf(gih + R2[lane * LDP + h], 0.0f) * w2h;
      s1 += fmaxf(gih + R2[(lane + 32) * LDP + h], 0.0f) * w2h;
    }
    gv[2 * ii + 0] = 1.0f / (1.0f + __expf(-(s0 + gb2)));
    gv[2 * ii + 1] = 1.0f / (1.0f + __expf(-(s1 + gb2)));
  }
  // gates overwrite gi in place: each wave reads/writes only its own 16 rows,
  // and per-lane all reads precede all writes (LDS is in-order per wave).
#pragma unroll
  for (int ii = 0; ii < 16; ++ii) {
    R1[(rowBase + ii) * LDP + lane]      = gv[2 * ii + 0];
    R1[(rowBase + ii) * LDP + lane + 32] = gv[2 * ii + 1];
  }
  __syncthreads();                                  // all waves done with gj

  // ---- msg = relu(enc @ msg_w1 + msg_b1) @ msg_w2 + msg_b2 ----
  copy_w_T(R2, msg_w1, 0, tid);                     // over dead gj
  __syncthreads();
  init_acc(acc, msg_b1, l16);
  gemm_accT(R0, LDP, R2, LDP, H_DIM, acc, lane, rowBase);
  __syncthreads();                                  // done reading enc & mw1T
  store_stripe(R0, LDP, acc, lane, rowBase, /*relu=*/true);   // tmp over enc
  copy_w_T(R2, msg_w2, 0, tid);                               // mw2T over mw1T
  __syncthreads();
  init_acc(acc, msg_b2, l16);
  gemm_accT(R0, LDP, R2, LDP, H_DIM, acc, lane, rowBase);
  __syncthreads();                                  // done reading mw2T
  store_stripeT(R2, LDP, acc, lane, rowBase);       // msgT over mw2T (B of P6)
  __syncthreads();

  // ---- out = gates @ msg ----
  init_acc(acc, nullptr, l16);
  gemm_accT(R1, LDP, R2, LDP, N_DIM, acc, lane, rowBase);

  float* outb = out + (size_t)b * (N_DIM * H_DIM);
  const int half = lane >> 4;
#pragma unroll
  for (int t = 0; t < 4; ++t)
#pragma unroll
    for (int r = 0; r < 8; ++r)
      outb[(rowBase + r + 8 * half) * H_DIM + t * 16 + l16] = acc[t][r];
}

extern "C" void kernel_launch(void* const* d_in, const int* in_sizes, int n_in,
                              void* d_out, int out_size, void* d_ws, size_t ws_size,
                              hipStream_t stream) {
  const float* obs     = (const float*)d_in[0];
  const float* enc_w   = (const float*)d_in[1];
  const float* enc_b   = (const float*)d_in[2];
  const float* gate_w1 = (const float*)d_in[3];
  const float* gate_b1 = (const float*)d_in[4];
  const float* gate_w2 = (const float*)d_in[5];
  const float* gate_b2 = (const float*)d_in[6];
  const float* msg_w1  = (const float*)d_in[7];
  const float* msg_b1  = (const float*)d_in[8];
  const float* msg_w2  = (const float*)d_in[9];
  const float* msg_b2  = (const float*)d_in[10];
  float* out = (float*)d_out;

  gated_comm_kernel<<<BATCH, 128, 0, stream>>>(
      obs, enc_w, enc_b, gate_w1, gate_b1, gate_w2, gate_b2,
      msg_w1, msg_b1, msg_w2, msg_b2, out);
}